// StyleBlock_72078141161950
// MI455X (gfx1250) — compile-verified
//
#include <hip/hip_runtime.h>
#include <math.h>

typedef float v2f __attribute__((ext_vector_type(2)));
typedef float v8f __attribute__((ext_vector_type(8)));
typedef int   v4i_b128 __attribute__((vector_size(16)));   // matches builtin param

// Problem constants (from reference setup_inputs)
#define BB   16
#define CIN  512
#define COUT 512
#define HH   32
#define WW   32
#define HWP  1024      // H*W pixels per image
#define LL   512

// d_ws layout (floats). Total = 2,637,824 floats = ~10.6 MB of scratch.
#define WS_STYLE 0                         // 16*512
#define WS_DEMOD 8192                      // 16*512
#define WS_WSQ   16384                     // 512*512
#define WS_WT    278528                    // 9*512*512  (wt[k][o][i], contiguous in i)

#if __has_builtin(__builtin_amdgcn_global_load_async_to_lds_b128)
#define USE_ASYNC_LDS 1
#else
#define USE_ASYNC_LDS 0
#endif

__device__ __forceinline__ void wait_asynccnt0() {
#if __has_builtin(__builtin_amdgcn_s_wait_asynccnt)
  __builtin_amdgcn_s_wait_asynccnt(0);
#else
  asm volatile("s_wait_asynccnt 0x0" ::: "memory");
#endif
}

// ---------------------------------------------------------------------------
// style[b,i] = sum_l latent[b,l] * affine_w[i,l] * sqrt(2/512) + affine_b[i]
// ---------------------------------------------------------------------------
__global__ void style_kernel(const float* __restrict__ latent,
                             const float* __restrict__ aw,
                             const float* __restrict__ ab,
                             float* __restrict__ style) {
  int t = blockIdx.x * blockDim.x + threadIdx.x;   // 0..8191
  int b = t >> 9, i = t & 511;
  const float* lrow = latent + b * LL;
  const float* wrow = aw + i * LL;
  float acc = 0.f;
#pragma unroll 8
  for (int l = 0; l < LL; ++l) acc += lrow[l] * wrow[l];
  style[t] = acc * 0.0625f + ab[i];                // sqrt(2/512) == 0.0625 exactly
}

// ---------------------------------------------------------------------------
// wt[k][o][i] = weight[o][i][k]  (GEMM-friendly, contiguous in i)
// wsq[o][i]   = sum_k weight[o][i][k]^2
// ---------------------------------------------------------------------------
__global__ void prep_weight_kernel(const float* __restrict__ w,
                                   float* __restrict__ wt,
                                   float* __restrict__ wsq) {
  int t = blockIdx.x * blockDim.x + threadIdx.x;   // 0..262143 == o*512+i
  int o = t >> 9, i = t & 511;
  const float* src = w + t * 9;
  float s = 0.f;
#pragma unroll
  for (int k = 0; k < 9; ++k) {
    float v = src[k];
    s += v * v;
    wt[(k * COUT + o) * CIN + i] = v;
  }
  wsq[t] = s;
}

// ---------------------------------------------------------------------------
// demod[b,o] = rsqrt( sum_i style[b,i]^2 * wsq[o,i] + 1e-8 )
// ---------------------------------------------------------------------------
__global__ void demod_kernel(const float* __restrict__ style,
                             const float* __restrict__ wsq,
                             float* __restrict__ demod) {
  int t = blockIdx.x * blockDim.x + threadIdx.x;   // 0..8191
  int b = t >> 9, o = t & 511;
  const float* srow = style + b * CIN;
  const float* qrow = wsq + o * CIN;
  float s = 0.f;
#pragma unroll 8
  for (int i = 0; i < CIN; ++i) {
    float st = srow[i];
    s += st * st * qrow[i];
  }
  demod[t] = rsqrtf(s + 1e-8f);
}

// ---------------------------------------------------------------------------
// Main fused conv: implicit GEMM (9 shifted GEMMs, register-accumulated)
// out[b,o,p] = leaky( demod[b,o]*sum_{k,i} wt[k][o][i]*style[b,i]*x_shift
//                     + nw*noise[b,p] + bias[o] , 0.2 )
// Block: 256 thr = 8 waves. Block tile 128(M=Cout) x 128(N=pixels), BK=32.
// Wave grid 4(M) x 2(N); each wave: 2x4 = 8 WMMA f32 16x16 C-tiles.
// A tile staged via GLOBAL_LOAD_ASYNC_TO_LDS_B128 (ASYNCcnt path) when
// available; B tile staged manually (style-modulate + shift + zero-pad).
// ---------------------------------------------------------------------------
__global__ void __launch_bounds__(256)
conv_mod_kernel(const float* __restrict__ x,
                const float* __restrict__ wt,
                const float* __restrict__ style,
                const float* __restrict__ demod,
                const float* __restrict__ bias,
                const float* __restrict__ noise,
                const float* __restrict__ nwp,
                float* __restrict__ out) {
  // pad rows to 36 floats: 8B-aligned v2f reads; 36*d==2 (mod 64) has no
  // solution -> conflict-free fragment reads across both lane halves.
  __shared__ float As[128][36];   // A tile: [M][K]   (K = 32)
  __shared__ float Bs[128][36];   // B tile transposed: [N][K]
  __shared__ float st_sh[CIN];    // style row for this batch

  const int tid = threadIdx.x;
  const int b   = blockIdx.z;
  const int o0  = blockIdx.y * 128;   // Cout tile base
  const int p0  = blockIdx.x * 128;   // pixel tile base

  st_sh[tid]       = style[b * CIN + tid];
  st_sh[tid + 256] = style[b * CIN + tid + 256];

  const int wv = tid >> 5, lane = tid & 31;
  const int wm = wv & 3;              // wave M position (0..3) -> 32 rows each
  const int wn = wv >> 2;             // wave N position (0..1) -> 64 cols each
  const int half = lane >> 4, ln = lane & 15;

  v8f acc[2][4];
#pragma unroll
  for (int mt = 0; mt < 2; ++mt)
#pragma unroll
    for (int nt = 0; nt < 4; ++nt)
      acc[mt][nt] = (v8f){0.f, 0.f, 0.f, 0.f, 0.f, 0.f, 0.f, 0.f};

  // cooperative-fill index maps (BK = 32)
  const int am = tid >> 1;            // A: row 0..127
  const int ac = (tid & 1) * 16;      // A: col base 0 or 16 (16 floats each)
  const int bk = tid >> 3;            // B: k index 0..31
  const int bp = (tid & 7) * 16;      // B: pixel base (16 consecutive)

  for (int kidx = 0; kidx < 9; ++kidx) {
    const int dh = kidx / 3 - 1;
    const int dw = kidx % 3 - 1;
    const float* wtk = wt + (size_t)(kidx * COUT + o0) * CIN;

    for (int i0 = 0; i0 < CIN; i0 += 32) {
      __syncthreads();
      // ---- stage A tile: 128x32 weights ----
      const float* gsrc = wtk + am * CIN + i0 + ac;
#if USE_ASYNC_LDS
      // CDNA5 async global->LDS copy (per-lane 16B), tracked by ASYNCcnt.
#pragma unroll
      for (int c = 0; c < 4; ++c)
        __builtin_amdgcn_global_load_async_to_lds_b128(
            (__attribute__((address_space(1))) v4i_b128*)(gsrc + c * 4),
            (__attribute__((address_space(3))) v4i_b128*)&As[am][ac + c * 4],
            0, 0);
#else
#pragma unroll
      for (int c = 0; c < 4; ++c)
        *(float4*)&As[am][ac + c * 4] = *(const float4*)(gsrc + c * 4);
#endif

      // ---- stage B tile: shifted+zero-padded input, modulated by style,
      //      stored transposed [pixel][k] ----
      const int ch = i0 + bk;
      const float sf = st_sh[ch];
      const float* xc = x + (size_t)(b * CIN + ch) * HWP;
#pragma unroll
      for (int j = 0; j < 16; ++j) {
        const int pp = p0 + bp + j;
        const int h = (pp >> 5) + dh;
        const int w = (pp & 31) + dw;
        float v = 0.f;
        if ((unsigned)h < (unsigned)HH && (unsigned)w < (unsigned)WW)
          v = xc[(h << 5) + w];
        Bs[bp + j][bk] = v * sf;
      }
#if USE_ASYNC_LDS
      wait_asynccnt0();               // A-tile async writes visible
#endif
      __syncthreads();

      // ---- 8 K-steps of 4: 8 v_wmma_f32_16x16x4_f32 each ----
#pragma unroll
      for (int k4 = 0; k4 < 8; ++k4) {
        const int kc = k4 * 4 + half * 2;  // lanes 0-15: K pair {0,1}; 16-31: {2,3}
        v2f afr[2], bfr[4];
#pragma unroll
        for (int mt = 0; mt < 2; ++mt)
          afr[mt] = *(const v2f*)&As[wm * 32 + mt * 16 + ln][kc];
#pragma unroll
        for (int nt = 0; nt < 4; ++nt)
          bfr[nt] = *(const v2f*)&Bs[wn * 64 + nt * 16 + ln][kc];
#pragma unroll
        for (int mt = 0; mt < 2; ++mt)
#pragma unroll
          for (int nt = 0; nt < 4; ++nt)
            acc[mt][nt] = __builtin_amdgcn_wmma_f32_16x16x4_f32(
                false, afr[mt], false, bfr[nt], (short)0, acc[mt][nt],
                false, false);
      }
    }
  }

  // ---- epilogue: demod scale + noise + bias + LeakyReLU(0.2) ----
  const float nw = *nwp;
#pragma unroll
  for (int mt = 0; mt < 2; ++mt) {
#pragma unroll
    for (int nt = 0; nt < 4; ++nt) {
      const int pg = p0 + wn * 64 + nt * 16 + ln;
      const float nz = nw * noise[b * HWP + pg];
      v8f c = acc[mt][nt];
#pragma unroll
      for (int r = 0; r < 8; ++r) {
        const int o = o0 + wm * 32 + mt * 16 + half * 8 + r;  // C layout: VGPR r -> M=r / r+8
        const float v = c[r] * demod[b * COUT + o] + nz + bias[o];
        out[(size_t)(b * COUT + o) * HWP + pg] = (v >= 0.f) ? v : 0.2f * v;
      }
    }
  }
}

// ---------------------------------------------------------------------------
extern "C" void kernel_launch(void* const* d_in, const int* in_sizes, int n_in,
                              void* d_out, int out_size, void* d_ws, size_t ws_size,
                              hipStream_t stream) {
  (void)in_sizes; (void)n_in; (void)out_size; (void)ws_size;
  const float* x        = (const float*)d_in[0];  // (16,512,32,32)
  const float* latent   = (const float*)d_in[1];  // (16,512)
  const float* weight   = (const float*)d_in[2];  // (512,512,3,3)
  const float* bias     = (const float*)d_in[3];  // (512,)
  const float* affine_w = (const float*)d_in[4];  // (512,512)
  const float* affine_b = (const float*)d_in[5];  // (512,)
  const float* noise    = (const float*)d_in[6];  // (16,1,32,32)
  const float* nwp      = (const float*)d_in[7];  // scalar
  float* out = (float*)d_out;
  float* ws  = (float*)d_ws;

  float* style = ws + WS_STYLE;
  float* demod = ws + WS_DEMOD;
  float* wsq   = ws + WS_WSQ;
  float* wt    = ws + WS_WT;

  style_kernel<<<32, 256, 0, stream>>>(latent, affine_w, affine_b, style);
  prep_weight_kernel<<<1024, 256, 0, stream>>>(weight, wt, wsq);
  demod_kernel<<<32, 256, 0, stream>>>(style, wsq, demod);

  dim3 grid(HWP / 128, COUT / 128, BB);   // (8, 4, 16) = 512 blocks
  conv_mod_kernel<<<grid, 256, 0, stream>>>(x, wt, style, demod, bias, noise,
                                            nwp, out);
}